// LearningModel_5901285064970
// MI455X (gfx1250) — compile-verified
//
#include <hip/hip_runtime.h>
#include <hip/hip_bf16.h>

// ---------------- problem constants (from reference) ----------------
#define N_INIT 20000
#define LVLS   32
#define P      1024
#define D      64
#define R      64
#define DE     128
#define N_TOT  (N_INIT + LVLS * P)   // 52768
#define NTILES (N_TOT / 16)          // 3298 (exact)

#define LVL_THREADS  128
#define EVAL_THREADS 256
#define EVAL_BLOCKS  416

// padded LDS strides (floats) to avoid bank conflicts (stride % 64 == 4)
// NOTE: W_S and W1_S are produced in hardware by the TDM pad feature.
#define PE_S 132   // 16 x 128 pe tile
#define W_S  68    // 128 x 64 rule weights (TDM pad: +4 dw every 64 dw)
#define X_S  68    // 16 x 64 store tile
#define W1_S 132   // 64 x 128 W1          (TDM pad: +4 dw every 128 dw)

typedef __attribute__((ext_vector_type(2))) float v2f;
typedef __attribute__((ext_vector_type(8))) float v8f;
typedef __attribute__((ext_vector_type(4))) unsigned int v4u;
typedef __attribute__((ext_vector_type(4))) int v4i;
typedef __attribute__((ext_vector_type(8))) int v8i;

// ---- TDM: issue a 1-D tensor_load_to_lds of `ndw` dwords from `gptr` to LDS
// offset `lds_off`, inserting pad dwords after every 2^(log2_interval_dw+1) dwords
// (reproduces the padded LDS row strides in hardware).
// Descriptor layout per CDNA5 ISA.
// 6-arg builtin flavor (clang-23 / therock-10.0 headers).
__device__ __forceinline__ void tdm_load_1d(const void* gptr, unsigned lds_off,
                                            unsigned ndw, unsigned log2_interval_dw,
                                            unsigned pad_dw_field) {
    unsigned long long ga = (unsigned long long)gptr;
    v4u g0;
    g0.x = 1u;                                   // count=1, is_restore=0, gather off
    g0.y = lds_off;                              // lds_addr (bytes)
    g0.z = (unsigned)(ga & 0xFFFFFFFFu);         // global_addr[31:0]
    g0.w = (unsigned)((ga >> 32) & 0x01FFFFFFu)  // global_addr[56:32]
         | (2u << 30);                           // type=2 ("image")
    v8i g1;
    // d0: wg_mask=0 | data_size=4B(2)<<16 | pad_enable<<20 | interval<<22 | amount<<25
    g1[0] = (int)((2u << 16) | (1u << 20) | (log2_interval_dw << 22) | (pad_dw_field << 25));
    g1[1] = (int)((ndw & 0xFFFFu) << 16);        // tensor_dim0[15:0]  (bits 63:48)
    g1[2] = (int)(((ndw >> 16) & 0xFFFFu)        // tensor_dim0[31:16] (bits 79:64)
         | (1u << 16));                          // tensor_dim1 = 1    (bits 111:80)
    g1[3] = (int)((ndw & 0xFFFFu) << 16);        // tile_dim0 = ndw    (bits 127:112)
    g1[4] = 1;                                   // tile_dim1 = 1, tile_dim2 = 0
    g1[5] = (int)ndw;                            // tensor_dim0_stride[31:0]
    g1[6] = 0;
    g1[7] = 0;
    v4i z4 = {0, 0, 0, 0};
    v8i z8 = {0, 0, 0, 0, 0, 0, 0, 0};
    __builtin_amdgcn_tensor_load_to_lds(g0, g1, z4, z4, z8, 0);
}

// ---------------- init: store[:N_INIT] = init_table[thax] ----------------
__global__ void k_init(const int* __restrict__ thax,
                       const float* __restrict__ init_table,
                       float* __restrict__ store) {
    int idx = blockIdx.x * blockDim.x + threadIdx.x;
    if (idx < N_INIT * D) {
        int node = idx >> 6, j = idx & 63;
        store[idx] = init_table[thax[node] * D + j];
    }
}

// ---------------- bucket: counting-sort nodes by rule, per level ----------------
__global__ void k_bucket(const int* __restrict__ rules,
                         int* __restrict__ order,
                         int* __restrict__ offsets) {
    __shared__ int cnt[R + 1];
    __shared__ int cur[R];
    const int l = blockIdx.x;
    const int t = threadIdx.x;
    if (t <= R) cnt[t] = 0;
    __syncthreads();
    for (int n = t; n < P; n += blockDim.x)
        atomicAdd(&cnt[rules[l * P + n]], 1);
    __syncthreads();
    if (t == 0) {
        int acc = 0;
        for (int r = 0; r < R; ++r) { int c = cnt[r]; cnt[r] = acc; cur[r] = acc; acc += c; }
        cnt[R] = acc;  // == P
        for (int r = 0; r <= R; ++r) offsets[l * (R + 1) + r] = cnt[r];
    }
    __syncthreads();
    for (int n = t; n < P; n += blockDim.x) {
        int r = rules[l * P + n];
        int pos = atomicAdd(&cur[r], 1);
        order[l * P + pos] = n;
    }
}

// ---------------- level: rule-grouped [16x128]@[128x64] f32 WMMA GEMMs ----------------
__global__ void k_level(const float* __restrict__ W_rule,
                        const float* __restrict__ b_rule,
                        const int* __restrict__ parents,
                        const int* __restrict__ order,
                        const int* __restrict__ offsets,
                        float* __restrict__ store, int l) {
    __shared__ float s_w[2 * D * W_S];   // 128 x 64 (TDM-padded to stride 68)
    __shared__ float s_pe[16 * PE_S];    // 16 x 128 (padded)
    __shared__ int   s_nodes[16];

    const int r    = blockIdx.x;
    const int beg  = offsets[l * (R + 1) + r];
    const int cnt  = offsets[l * (R + 1) + r + 1] - beg;
    if (cnt == 0) return;

    const int t = threadIdx.x;
    const int lane  = t & 31;
    const int wave  = t >> 5;            // 0..3 : N-tile of 16 output cols
    const int col   = lane & 15;         // A row (M) / B,D col (N) per WMMA layout
    const int khalf = (lane >> 4) << 1;  // 0 or 2: K pair selector
    const int half8 = (lane >> 4) << 3;  // 0 or 8: D row offset

    // --- TDM: stage W_rule[r] (8192 dwords) into LDS, +4dw pad every 64dw ---
    if (wave == 0) {
        tdm_load_1d(W_rule + (size_t)r * 2 * D * D,
                    (unsigned)(size_t)&s_w[0],
                    2 * D * D, /*interval 64dw*/5, /*pad 4dw*/3);
        __builtin_amdgcn_s_wait_tensorcnt(0);
    }

    const float bias = b_rule[r * D + wave * 16 + col];

    for (int g = 0; g < cnt; g += 16) {
        __syncthreads();   // also orders TDM-filled s_w with consumers
        if (t < 16) s_nodes[t] = (g + t < cnt) ? order[l * P + beg + g + t] : -1;
        __syncthreads();
        // gather 16 nodes' parent embeddings -> [16][128]
        for (int idx = t; idx < 16 * 2 * D; idx += LVL_THREADS) {
            int m = idx >> 7, i = idx & 127;
            int node = s_nodes[m];
            float v = 0.f;
            if (node >= 0) {
                int p = parents[((size_t)l * P + node) * 2 + (i >> 6)];
                v = store[(size_t)p * D + (i & 63)];
            }
            s_pe[m * PE_S + i] = v;
        }
        __syncthreads();

        v8f acc = {};
        #pragma unroll
        for (int kk = 0; kk < 2 * D; kk += 4) {
            v2f a, b;
            a.x = s_pe[col * PE_S + kk + khalf];
            a.y = s_pe[col * PE_S + kk + khalf + 1];
            b.x = s_w[(kk + khalf) * W_S + wave * 16 + col];
            b.y = s_w[(kk + khalf + 1) * W_S + wave * 16 + col];
            acc = __builtin_amdgcn_wmma_f32_16x16x4_f32(
                false, a, false, b, (short)0, acc, false, false);
        }
        #pragma unroll
        for (int i = 0; i < 8; ++i) {
            int m = i + half8;
            int node = s_nodes[m];
            if (node >= 0) {
                float h = fmaxf(acc[i] + bias, 0.f);
                store[(size_t)(N_INIT + l * P + node) * D + wave * 16 + col] = h;
            }
        }
    }
}

// ---------------- eval: [N_TOT x 64]@[64 x 128] WMMA + weighted BCE partials ----------------
__global__ void k_eval(const float* __restrict__ store,
                       const float* __restrict__ W1,
                       const float* __restrict__ b1,
                       const float* __restrict__ w2,
                       const float* __restrict__ b2,
                       const int* __restrict__ sel_mask,
                       const int* __restrict__ good_mask,
                       float* __restrict__ partials) {
    __shared__ float s_w[D * W1_S];    // 64 x 128 (TDM-padded to stride 132)
    __shared__ float s_x[16 * X_S];    // 16 x 64  (padded)
    __shared__ float s_part[8 * 16];   // [wave][row]
    __shared__ float s_red[16 * 6];

    const int t = threadIdx.x;
    const int lane  = t & 31;
    const int wave  = t >> 5;            // 0..7 : N-tile of DE=128
    const int col   = lane & 15;
    const int khalf = (lane >> 4) << 1;
    const int half8 = (lane >> 4) << 3;
    const int ncol  = wave * 16 + col;

    // --- TDM: stage W1 (8192 dwords) into LDS, +4dw pad every 128dw ---
    if (wave == 0) {
        tdm_load_1d(W1, (unsigned)(size_t)&s_w[0],
                    D * DE, /*interval 128dw*/6, /*pad 4dw*/3);
        __builtin_amdgcn_s_wait_tensorcnt(0);
    }

    const float bias1 = b1[ncol];
    const float wv2   = w2[ncol];
    const float b2v   = b2[0];

    float rs[6] = {0.f, 0.f, 0.f, 0.f, 0.f, 0.f};

    for (int tile = blockIdx.x; tile < NTILES; tile += gridDim.x) {
        __syncthreads();   // also orders TDM-filled s_w with consumers
        for (int idx = t; idx < 16 * D; idx += EVAL_THREADS) {
            int m = idx >> 6, k = idx & 63;
            s_x[m * X_S + k] = store[(size_t)(tile * 16 + m) * D + k];
        }
        __syncthreads();

        v8f acc = {};
        #pragma unroll
        for (int kk = 0; kk < D; kk += 4) {
            v2f a, b;
            a.x = s_x[col * X_S + kk + khalf];
            a.y = s_x[col * X_S + kk + khalf + 1];
            b.x = s_w[(kk + khalf) * W1_S + ncol];
            b.y = s_w[(kk + khalf + 1) * W1_S + ncol];
            acc = __builtin_amdgcn_wmma_f32_16x16x4_f32(
                false, a, false, b, (short)0, acc, false, false);
        }
        // h = relu(acc + b1); per-row partial dot with w2 over this wave's 16 cols
        #pragma unroll
        for (int i = 0; i < 8; ++i) {
            float s = fmaxf(acc[i] + bias1, 0.f) * wv2;
            s += __shfl_xor(s, 1, 32);
            s += __shfl_xor(s, 2, 32);
            s += __shfl_xor(s, 4, 32);
            s += __shfl_xor(s, 8, 32);
            if (col == 0) s_part[wave * 16 + i + half8] = s;
        }
        __syncthreads();
        if (t < 16) {
            float lg = b2v;
            #pragma unroll
            for (int w = 0; w < 8; ++w) lg += s_part[w * 16 + t];
            int node = tile * 16 + t;
            bool sel  = sel_mask[node] > 0;
            bool good = (good_mask[node] > 0) && sel;
            float x  = lg;
            float sp = log1pf(expf(-fabsf(x))) + fmaxf(x, 0.f);  // softplus, stable
            float bce = sp - (good ? x : 0.f);
            if (good)            { rs[0] += bce; rs[4] += 1.f; }
            if (sel && !good)    { rs[1] += bce; }
            if (good && x >= 0.f){ rs[2] += 1.f; }
            if (sel && !good && x < 0.f) { rs[3] += 1.f; }
            if (sel)             { rs[5] += 1.f; }
        }
    }
    // deterministic block-level reduce of the 16 accumulator threads
    __syncthreads();
    if (t < 16) {
        #pragma unroll
        for (int j = 0; j < 6; ++j) s_red[t * 6 + j] = rs[j];
    }
    __syncthreads();
    if (t == 0) {
        float out[6] = {0.f, 0.f, 0.f, 0.f, 0.f, 0.f};
        for (int m = 0; m < 16; ++m)
            for (int j = 0; j < 6; ++j) out[j] += s_red[m * 6 + j];
        for (int j = 0; j < 6; ++j) partials[blockIdx.x * 6 + j] = out[j];
    }
}

// ---------------- finalize: deterministic serial reduction + loss math ----------------
__global__ void k_final(const float* __restrict__ partials, float* __restrict__ out) {
    if (threadIdx.x != 0 || blockIdx.x != 0) return;
    float S[6] = {0.f, 0.f, 0.f, 0.f, 0.f, 0.f};
    for (int b = 0; b < EVAL_BLOCKS; ++b)
        for (int j = 0; j < 6; ++j) S[j] += partials[b * 6 + j];
    float n_good = S[4];
    float n_sel  = S[5];
    float n_neg  = n_sel - n_good;
    float pos_w = (n_good > 0.f) ? 0.85f / fmaxf(n_good, 1.f) : 1.f;
    float neg_w = (n_neg > 0.f) ? 0.15f / fmaxf(n_neg, 1.f) : 1.f;
    out[0] = pos_w * S[0] + neg_w * S[1];
    out[1] = (n_good > 0.f) ? S[2] / fmaxf(n_good, 1.f) : 1.f;
    out[2] = (n_neg > 0.f) ? S[3] / fmaxf(n_neg, 1.f) : 1.f;
}

// ---------------- host: launch pipeline on stream ----------------
extern "C" void kernel_launch(void* const* d_in, const int* in_sizes, int n_in,
                              void* d_out, int out_size, void* d_ws, size_t ws_size,
                              hipStream_t stream) {
    const int*   thax       = (const int*)  d_in[0];
    const int*   parents    = (const int*)  d_in[1];
    const int*   rules      = (const int*)  d_in[2];
    const int*   sel_mask   = (const int*)  d_in[3];
    const int*   good_mask  = (const int*)  d_in[4];
    const float* init_table = (const float*)d_in[5];
    const float* W_rule     = (const float*)d_in[6];
    const float* b_rule     = (const float*)d_in[7];
    const float* W1         = (const float*)d_in[8];
    const float* b1         = (const float*)d_in[9];
    const float* w2         = (const float*)d_in[10];
    const float* b2         = (const float*)d_in[11];

    char* ws = (char*)d_ws;
    float* store = (float*)ws;
    size_t off = (size_t)N_TOT * D * sizeof(float);           // 13.5 MB
    int* order = (int*)(ws + off);   off += (size_t)LVLS * P * sizeof(int);
    int* offsets = (int*)(ws + off); off += (size_t)LVLS * (R + 1) * sizeof(int);
    float* partials = (float*)(ws + off);

    k_init<<<(N_INIT * D + 255) / 256, 256, 0, stream>>>(thax, init_table, store);
    k_bucket<<<LVLS, 256, 0, stream>>>(rules, order, offsets);
    for (int l = 0; l < LVLS; ++l)
        k_level<<<R, LVL_THREADS, 0, stream>>>(W_rule, b_rule, parents, order,
                                               offsets, store, l);
    k_eval<<<EVAL_BLOCKS, EVAL_THREADS, 0, stream>>>(store, W1, b1, w2, b2,
                                                     sel_mask, good_mask, partials);
    k_final<<<1, 1, 0, stream>>>(partials, (float*)d_out);
}